// PolyGCLModel_71184787964541
// MI455X (gfx1250) — compile-verified
//
#include <hip/hip_runtime.h>
#include <hip/hip_bf16.h>

#define IN_DIM 512
#define H_DIM 128
#define OUT_DIM 128
#define BN_EPS 1e-5f

typedef __attribute__((ext_vector_type(2))) float v2f;
typedef __attribute__((ext_vector_type(4))) float v4f;
typedef __attribute__((ext_vector_type(8))) float v8f;

// ---------------------------------------------------------------------------
// small elementwise / setup kernels
// ---------------------------------------------------------------------------
__global__ void zero_kernel(float* __restrict__ p, long long n) {
    long long i = (long long)blockIdx.x * blockDim.x + threadIdx.x;
    long long s = (long long)gridDim.x * blockDim.x;
    for (; i < n; i += s) p[i] = 0.0f;
}

__global__ void deg_accum_kernel(const long long* __restrict__ dst,
                                 float* __restrict__ deg, int E) {
    int i = blockIdx.x * blockDim.x + threadIdx.x;
    int s = gridDim.x * blockDim.x;
    for (; i < E; i += s) atomicAdd(&deg[dst[i]], 1.0f);
}

__global__ void deg_finalize_kernel(float* __restrict__ deg,
                                    float* __restrict__ invdeg, int N) {
    int i = blockIdx.x * blockDim.x + threadIdx.x;
    if (i < N) {
        float d = deg[i] + 1.0f;   // self loop
        deg[i] = d;
        invdeg[i] = 1.0f / d;
    }
}

__global__ void edge_w_kernel(const long long* __restrict__ src,
                              const long long* __restrict__ dst,
                              const float* __restrict__ deg,
                              float* __restrict__ w, int E) {
    int i = blockIdx.x * blockDim.x + threadIdx.x;
    int s = gridDim.x * blockDim.x;
    for (; i < E; i += s) w[i] = rsqrtf(deg[src[i]] * deg[dst[i]]);
}

__global__ void init_acc_kernel(const float* __restrict__ hk,
                                const float* __restrict__ gammas,
                                float* __restrict__ acc, long long n) {
    const float g0 = gammas[0];
    long long i = (long long)blockIdx.x * blockDim.x + threadIdx.x;
    long long s = (long long)gridDim.x * blockDim.x;
    for (; i < n; i += s) acc[i] = g0 * hk[i];
}

// msg = hk / deg   (self-loop term, also serves as the zero-init of msg)
__global__ void prep_kernel(const float* __restrict__ hk,
                            const float* __restrict__ invdeg,
                            float* __restrict__ msg, long long n) {
    long long i = (long long)blockIdx.x * blockDim.x + threadIdx.x;
    long long s = (long long)gridDim.x * blockDim.x;
    for (; i < n; i += s) msg[i] = hk[i] * invdeg[i >> 7];   // H = 128
}

// one edge per wave32: 128-bit gather of hk[src], 4x f32 atomic add into msg[dst]
__global__ __launch_bounds__(256)
void scatter_kernel(const long long* __restrict__ src,
                    const long long* __restrict__ dst,
                    const float* __restrict__ w,
                    const float* __restrict__ hk,
                    float* __restrict__ msg, int E) {
    const int lane = threadIdx.x & 31;
    int gw = (blockIdx.x * blockDim.x + threadIdx.x) >> 5;
    const int nw = (gridDim.x * blockDim.x) >> 5;
    for (int e = gw; e < E; e += nw) {
        long long s = src[e];
        long long d = dst[e];
        float we = w[e];
        if (e + nw < E) {   // prefetch next gather row into caches
            long long sn = src[e + nw];
            __builtin_prefetch(hk + sn * H_DIM + lane * 4, 0, 1);
        }
        v4f v = ((const v4f*)(hk + s * H_DIM))[lane];
        float* mp = msg + d * H_DIM + lane * 4;
        atomicAdd(mp + 0, we * v.x);
        atomicAdd(mp + 1, we * v.y);
        atomicAdd(mp + 2, we * v.z);
        atomicAdd(mp + 3, we * v.w);
    }
}

// hk = high_pass ? hk - msg : msg ;  acc += gamma_k * hk
__global__ void combine_kernel(float* __restrict__ hk,
                               const float* __restrict__ msg,
                               float* __restrict__ acc,
                               const float* __restrict__ gammas,
                               const int* __restrict__ hp,
                               int kidx, long long n) {
    const float g = gammas[kidx];
    const int high = hp[0];
    long long i = (long long)blockIdx.x * blockDim.x + threadIdx.x;
    long long s = (long long)gridDim.x * blockDim.x;
    for (; i < n; i += s) {
        float m = msg[i];
        float v = high ? (hk[i] - m) : m;
        hk[i] = v;
        acc[i] += g * v;
    }
}

// ---------------------------------------------------------------------------
// BatchNorm statistics
// ---------------------------------------------------------------------------
__global__ void bn_zero_kernel(float* __restrict__ sums) {
    sums[threadIdx.x] = 0.0f;   // launched with 256 threads (sum + sumsq)
}

__global__ __launch_bounds__(128)
void bn_stats_kernel(const float* __restrict__ acc,
                     float* __restrict__ sums, int N) {
    const int ch = threadIdx.x;          // channel per lane -> coalesced rows
    float s = 0.0f, s2 = 0.0f;
    for (int r = blockIdx.x; r < N; r += gridDim.x) {
        float v = acc[(long long)r * H_DIM + ch];
        s += v;
        s2 += v * v;
    }
    atomicAdd(&sums[ch], s);
    atomicAdd(&sums[H_DIM + ch], s2);
}

__global__ void bn_finalize_kernel(const float* __restrict__ sums,
                                   const float* __restrict__ bng,
                                   const float* __restrict__ bnb,
                                   float* __restrict__ scale,
                                   float* __restrict__ shift, int N) {
    int ch = threadIdx.x;                // 128 threads
    float invN = 1.0f / (float)N;
    float mu  = sums[ch] * invN;
    float var = sums[H_DIM + ch] * invN - mu * mu;
    float sc  = bng[ch] * rsqrtf(var + BN_EPS);
    scale[ch] = sc;
    shift[ch] = bnb[ch] - mu * sc;
}

// ---------------------------------------------------------------------------
// f32 WMMA GEMM: h[N,128] = x[N,512] @ W_enc[128,512]^T + b_enc
// block = 256 threads (8 waves); wave w owns column tile n0 = 16*w;
// x row-tile (16x512, 32KB) staged in LDS; W_enc rows stream from L2.
// V_WMMA_F32_16X16X4_F32 operand layout (ISA 7.12.2):
//   A (16x4): lanes 0-15 -> M=lane, K=k,k+1 ; lanes 16-31 -> M=lane-16, K=k+2,k+3
//   B (4x16): same with M->N ; D: vgpr r -> M = r (+8 for upper lanes), N = lane&15
// ---------------------------------------------------------------------------
__global__ __launch_bounds__(256)
void encoder_wmma_kernel(const float* __restrict__ x,
                         const float* __restrict__ Wenc,
                         const float* __restrict__ benc,
                         float* __restrict__ h, int N) {
    __shared__ float xs[16 * IN_DIM];            // 32 KB
    const int row0 = blockIdx.x * 16;
    {
        const int t = threadIdx.x;
        for (int i = 0; i < 8; ++i) {            // 2048 float4 / 256 threads
            int idx4 = t + i * 256;
            int r = idx4 >> 7;                   // 128 float4 per row
            int c4 = idx4 & 127;
            int row = row0 + r; if (row >= N) row = N - 1;
            ((v4f*)xs)[idx4] = ((const v4f*)(x + (long long)row * IN_DIM))[c4];
        }
    }
    __syncthreads();

    const int wave = threadIdx.x >> 5;           // 0..7 -> column tile
    const int lane = threadIdx.x & 31;
    const int n0   = wave * 16;
    const int half = lane >> 4;                  // 0 | 1
    const int mn   = lane & 15;
    const int koff = half * 2;

    v8f c = {};
    const float* arow = xs + mn * IN_DIM;
    const float* wrow = Wenc + (long long)(n0 + mn) * IN_DIM;
    for (int k = 0; k < IN_DIM; k += 4) {
        v2f a; a.x = arow[k + koff]; a.y = arow[k + koff + 1];
        v2f b; b.x = wrow[k + koff]; b.y = wrow[k + koff + 1];
        c = __builtin_amdgcn_wmma_f32_16x16x4_f32(false, a, false, b,
                                                  (short)0, c, false, false);
    }
    const float bias = benc[n0 + mn];
    for (int r = 0; r < 8; ++r) {
        int row = row0 + r + half * 8;
        if (row < N) h[(long long)row * H_DIM + n0 + mn] = c[r] + bias;
    }
}

// ---------------------------------------------------------------------------
// Head: z = acc*scale + shift (BN affine folded into LDS staging),
// y = z @ W_up^T + b_up, PReLU.  acc lives in `out`; each block reads only its
// own 16 rows (before the barrier) and overwrites them afterwards.
// ---------------------------------------------------------------------------
__global__ __launch_bounds__(256)
void head_wmma_kernel(float* out,
                      const float* __restrict__ scale,
                      const float* __restrict__ shift,
                      const float* __restrict__ Wup,
                      const float* __restrict__ bup,
                      const float* __restrict__ alpha_p,
                      int N) {
    __shared__ float zs[16 * H_DIM];             // 8 KB
    const int row0 = blockIdx.x * 16;
    {
        const int t = threadIdx.x;
        for (int i = 0; i < 8; ++i) {            // 2048 floats / 256 threads
            int idx = t + i * 256;
            int r = idx >> 7;
            int ch = idx & 127;
            int row = row0 + r; if (row >= N) row = N - 1;
            zs[idx] = out[(long long)row * H_DIM + ch] * scale[ch] + shift[ch];
        }
    }
    __syncthreads();

    const int wave = threadIdx.x >> 5;
    const int lane = threadIdx.x & 31;
    const int n0   = wave * 16;
    const int half = lane >> 4;
    const int mn   = lane & 15;
    const int koff = half * 2;

    v8f c = {};
    const float* arow = zs + mn * H_DIM;
    const float* wrow = Wup + (long long)(n0 + mn) * H_DIM;
    for (int k = 0; k < H_DIM; k += 4) {
        v2f a; a.x = arow[k + koff]; a.y = arow[k + koff + 1];
        v2f b; b.x = wrow[k + koff]; b.y = wrow[k + koff + 1];
        c = __builtin_amdgcn_wmma_f32_16x16x4_f32(false, a, false, b,
                                                  (short)0, c, false, false);
    }
    const float bias  = bup[n0 + mn];
    const float alpha = alpha_p[0];
    for (int r = 0; r < 8; ++r) {
        int row = row0 + r + half * 8;
        if (row < N) {
            float y = c[r] + bias;
            out[(long long)row * OUT_DIM + n0 + mn] = (y > 0.0f) ? y : alpha * y;
        }
    }
}

// ---------------------------------------------------------------------------
extern "C" void kernel_launch(void* const* d_in, const int* in_sizes, int n_in,
                              void* d_out, int out_size, void* d_ws, size_t ws_size,
                              hipStream_t stream) {
    const float*     x      = (const float*)d_in[0];
    const float*     Wenc   = (const float*)d_in[1];
    const float*     benc   = (const float*)d_in[2];
    const float*     gammas = (const float*)d_in[3];
    const float*     bng    = (const float*)d_in[4];
    const float*     bnb    = (const float*)d_in[5];
    const float*     Wup    = (const float*)d_in[6];
    const float*     bup    = (const float*)d_in[7];
    const float*     alpha  = (const float*)d_in[8];
    const long long* ei     = (const long long*)d_in[9];   // int64 edge_index
    const int*       hp     = (const int*)d_in[10];

    const int N = in_sizes[0] / IN_DIM;
    const int E = in_sizes[9] / 2;
    const int K = in_sizes[3] - 1;
    const long long NH = (long long)N * H_DIM;

    const long long* srcI = ei;
    const long long* dstI = ei + E;

    // workspace layout (floats): deg[N] invdeg[N] w[E] hk[N*H] msg[N*H] sums[256] scale[128] shift[128]
    float* ws     = (float*)d_ws;
    float* deg    = ws;
    float* invdeg = deg + N;
    float* wE     = invdeg + N;
    float* hk     = wE + E;
    float* msg    = hk + NH;
    float* sums   = msg + NH;
    float* scale  = sums + 256;
    float* shift  = scale + 128;
    float* acc    = (float*)d_out;       // reused: acc then final output (same shape)

    const int tileBlocks = (N + 15) / 16;

    zero_kernel<<<512, 256, 0, stream>>>(deg, N);
    deg_accum_kernel<<<2048, 256, 0, stream>>>(dstI, deg, E);
    deg_finalize_kernel<<<(N + 255) / 256, 256, 0, stream>>>(deg, invdeg, N);
    edge_w_kernel<<<2048, 256, 0, stream>>>(srcI, dstI, deg, wE, E);

    encoder_wmma_kernel<<<tileBlocks, 256, 0, stream>>>(x, Wenc, benc, hk, N);
    init_acc_kernel<<<2048, 256, 0, stream>>>(hk, gammas, acc, NH);

    for (int k = 1; k <= K; ++k) {
        prep_kernel<<<2048, 256, 0, stream>>>(hk, invdeg, msg, NH);
        scatter_kernel<<<25000, 256, 0, stream>>>(srcI, dstI, wE, hk, msg, E);
        combine_kernel<<<2048, 256, 0, stream>>>(hk, msg, acc, gammas, hp, k, NH);
    }

    bn_zero_kernel<<<1, 256, 0, stream>>>(sums);
    bn_stats_kernel<<<1024, 128, 0, stream>>>(acc, sums, N);
    bn_finalize_kernel<<<1, 128, 0, stream>>>(sums, bng, bnb, scale, shift, N);

    head_wmma_kernel<<<tileBlocks, 256, 0, stream>>>(acc, scale, shift, Wup, bup, alpha, N);
}